// WeightedRuleLayer_30511447671661
// MI455X (gfx1250) — compile-verified
//
#include <hip/hip_runtime.h>
#include <stdint.h>

// WeightedRuleLayer for MI455X (gfx1250, wave32).
//
//   out[r, :] = tanh( sum_p layer_values[idx[r,p], :] * weights[r, p, :] )
//   R = 250000, P = 8, D = 128
//
// Memory-bound (~1.16 GB HBM traffic -> ~50us at 23.3 TB/s).
// - CDNA5 async global->LDS copies (GLOBAL_LOAD_ASYNC_TO_LDS_B128 / ASYNCcnt)
//   stage the streaming per-rule weight block and the 8 gathered rows.
// - Weights are read-once: marked th:TH_LOAD_NT so the 1 GB stream does not
//   evict the 102 MB layer_values table from the 192 MB L2 (gathers stay
//   L2-resident at RT policy).
// - Indices / output are touch-once: nontemporal load/store.

#define D_DIM   128
#define PERIOD  8
#define WAVES_PER_BLOCK 8           // 256 threads = 8 wave32

typedef __attribute__((ext_vector_type(4))) float v4f;

__global__ __launch_bounds__(256)
void rule_layer_kernel(const float* __restrict__ layer_values,
                       const float* __restrict__ weights,
                       const int*   __restrict__ indices,
                       float*       __restrict__ out,
                       int n_rules)
{
    // Per wave: 4KB weights + 4KB gathered rows. Per block: 64KB (<320KB WGP LDS).
    __shared__ v4f smem[WAVES_PER_BLOCK * 2 * PERIOD * (D_DIM / 4)];

    const int lane = threadIdx.x & 31;
    const int wave = threadIdx.x >> 5;
    const int rule = blockIdx.x * WAVES_PER_BLOCK + wave;
    if (rule >= n_rules) return;          // wave-uniform guard

    v4f* sW = &smem[wave * (2 * PERIOD * (D_DIM / 4))];      // 256 x 16B
    v4f* sG = sW + PERIOD * (D_DIM / 4);                     // 256 x 16B

    // LDS byte address = low 32 bits of the flat (generic) address on amdgcn.
    const uint32_t ldsWBase = (uint32_t)(uintptr_t)sW;
    const uint32_t ldsGBase = (uint32_t)(uintptr_t)sG;

    const unsigned long long wBase  = (unsigned long long)(uintptr_t)weights;
    const unsigned long long lvBase = (unsigned long long)(uintptr_t)layer_values;

    const uint32_t laneByte   = (uint32_t)lane * 16u;                   // 16B per lane
    const uint32_t wRuleBytes = (uint32_t)rule * (PERIOD * D_DIM * 4u); // rule * 4096

    // ---- stage the 4KB per-rule weight block: 8 async b128 ops (512B each),
    //      non-temporal so the 1 GB weight stream bypasses L2 residency ----
#pragma unroll
    for (int p = 0; p < PERIOD; ++p) {
        uint32_t ldsOff = ldsWBase + (uint32_t)p * (D_DIM * 4u) + laneByte;
        uint32_t gOff   = wRuleBytes + (uint32_t)p * (D_DIM * 4u) + laneByte;
        asm volatile("global_load_async_to_lds_b128 %0, %1, %2 th:TH_LOAD_NT"
                     :: "v"(ldsOff), "v"(gOff), "s"(wBase) : "memory");
    }

    // ---- indices: lane L loads index (L&7); broadcast per-p with a shuffle.
    //      Touch-once stream: nontemporal. ----
    int myIdx = __builtin_nontemporal_load(&indices[rule * PERIOD + (lane & (PERIOD - 1))]);

    // ---- stage the 8 gathered rows (contiguous 512B each, L2-resident: RT) ----
#pragma unroll
    for (int p = 0; p < PERIOD; ++p) {
        int idx = __shfl(myIdx, p, 32);
        uint32_t ldsOff = ldsGBase + (uint32_t)p * (D_DIM * 4u) + laneByte;
        uint32_t gOff   = (uint32_t)idx * (D_DIM * 4u) + laneByte;
        asm volatile("global_load_async_to_lds_b128 %0, %1, %2"
                     :: "v"(ldsOff), "v"(gOff), "s"(lvBase) : "memory");
    }

    // Wait for all async LDS writes of this wave to land.
    asm volatile("s_wait_asynccnt 0" ::: "memory");

    // ---- Hadamard-multiply + reduce over P, from LDS (ds_load_b128) ----
    v4f acc = (v4f){0.f, 0.f, 0.f, 0.f};
#pragma unroll
    for (int p = 0; p < PERIOD; ++p) {
        v4f w = sW[p * (D_DIM / 4) + lane];
        v4f g = sG[p * (D_DIM / 4) + lane];
        acc.x = fmaf(g.x, w.x, acc.x);
        acc.y = fmaf(g.y, w.y, acc.y);
        acc.z = fmaf(g.z, w.z, acc.z);
        acc.w = fmaf(g.w, w.w, acc.w);
    }

    v4f r;
    r.x = tanhf(acc.x);
    r.y = tanhf(acc.y);
    r.z = tanhf(acc.z);
    r.w = tanhf(acc.w);

    // 512B coalesced, write-once -> nontemporal store (global_store_b128 NT)
    v4f* out4 = (v4f*)out;
    __builtin_nontemporal_store(r, &out4[(size_t)rule * (D_DIM / 4) + lane]);
}

extern "C" void kernel_launch(void* const* d_in, const int* in_sizes, int n_in,
                              void* d_out, int out_size, void* d_ws, size_t ws_size,
                              hipStream_t stream) {
    const float* layer_values = (const float*)d_in[0];
    const float* weights      = (const float*)d_in[1];
    const int*   indices      = (const int*)d_in[2];
    float*       out          = (float*)d_out;

    const int n_rules = out_size / D_DIM;                       // 250000
    const int blocks  = (n_rules + WAVES_PER_BLOCK - 1) / WAVES_PER_BLOCK;

    hipLaunchKernelGGL(rule_layer_kernel, dim3(blocks), dim3(256), 0, stream,
                       layer_values, weights, indices, out, n_rules);
}